// CrossAttentionFusion_39127152066813
// MI455X (gfx1250) — compile-verified
//
#include <hip/hip_runtime.h>
#include <hip/hip_bf16.h>

// ---------------------------------------------------------------------------
// CrossAttentionFusion for MI455X (gfx1250), fp32 end-to-end via
// V_WMMA_F32_16X16X4_F32 (exact fp32 matrix path, wave32).
//
// Math (seq_len==1 => softmax == 1):
//   attn = X @ (Wo @ Wv)^T + (Wo @ bv + bo)
//   fused_g = LN(g + attn_g2s),  fused_s = LN(s + attn_s2g)
//   gate = sigmoid([fused_g, fused_s] @ Wg^T + bg)
//   out  = gate*fused_g + (1-gate)*fused_s ; also output gate
//
// GEMM core: 256 threads = 8 waves; block tile 64(M) x 128(N); each wave owns
// a 32x32 sub-tile as 4 WMMA accumulators (2 A-frags x 2 B-frags, 1:1
// LDS-load:WMMA ratio). K staged through padded LDS in chunks of 32.
// ---------------------------------------------------------------------------

#define BDIM 32768
#define DDIM 1024
#define LN_EPS 1e-5f

typedef __attribute__((ext_vector_type(2))) float v2f;
typedef __attribute__((ext_vector_type(8))) float v8f;

#define KT 32
#define LDSS 36   // padded LDS row stride (dwords): 16B aligned, conflict-light

enum { EPI_STORE = 0, EPI_RESID_BIAS = 1, EPI_GATE = 2 };

static __device__ __forceinline__ v8f wmma4(v2f a, v2f b, v8f c) {
  return __builtin_amdgcn_wmma_f32_16x16x4_f32(
      /*neg_a=*/false, a, /*neg_b=*/false, b,
      /*c_mod=*/(short)0, c, /*reuse_a=*/false, /*reuse_b=*/false);
}

// ---------------------------------------------------------------------------
// Generic fp32 WMMA GEMM:  C[m,n] = sum_k A[m,k] * B'[k,n]  (+ epilogue)
//   B_TRANSPOSED=true :  B'[k,n] = Bm[n*ldb + k]   (X @ W^T, weights row-major)
//   B_TRANSPOSED=false:  B'[k,n] = Bm[k*ldb + n]   (plain X @ W)
// ---------------------------------------------------------------------------
template <int EPI, bool B_TRANSPOSED>
__global__ __launch_bounds__(256)
void gemm_f32_wmma(const float* __restrict__ A, int lda,
                   const float* __restrict__ Bm, int ldb,
                   float* __restrict__ C, int ldc, int K,
                   const float* __restrict__ resid, int ld_resid,
                   const float* __restrict__ bias,
                   const float* __restrict__ catbuf, int ldcat, int fs_off,
                   float* __restrict__ out_gate) {
  __shared__ float At[64 * LDSS];
  __shared__ float Bt[128 * LDSS];

  const int t    = threadIdx.x;
  const int lane = t & 31;
  const int wave = t >> 5;
  const int wm   = wave & 1;       // 0..1 -> 32-row sub-block
  const int wn   = wave >> 1;      // 0..3 -> 32-col sub-block
  const int row0 = blockIdx.y * 64;
  const int col0 = blockIdx.x * 128;

  const int m_l  = lane & 15;                 // M (for A) / N (for B) index
  const int koff = (lane < 16) ? 0 : 2;       // per ISA 16x16x4 f32 layout

  v8f acc[2][2];
#pragma unroll
  for (int i = 0; i < 2; ++i)
#pragma unroll
    for (int j = 0; j < 2; ++j)
      acc[i][j] = (v8f){0.f, 0.f, 0.f, 0.f, 0.f, 0.f, 0.f, 0.f};

  const float* ap0 = At + (wm * 32 + m_l) * LDSS + koff;
  const float* ap1 = ap0 + 16 * LDSS;
  const float* bp0 = Bt + (wn * 32 + m_l) * LDSS + koff;
  const float* bp1 = bp0 + 16 * LDSS;

  for (int kb = 0; kb < K; kb += KT) {
    __syncthreads();  // previous chunk fully consumed
    // --- stage A tile: 64 rows x KT cols, two float4 per thread ---
    {
      const int r  = t >> 3;
      const int kq = (t & 7) << 2;
#pragma unroll
      for (int i = 0; i < 2; ++i) {
        const int rr = r + 32 * i;
        const float4 v = *(const float4*)(A + (size_t)(row0 + rr) * lda + kb + kq);
        *(float4*)(At + rr * LDSS + kq) = v;
      }
    }
    // --- stage B tile as Bt[n][k]: 128 x KT ---
    if (B_TRANSPOSED) {
      const int r  = t >> 3;
      const int kq = (t & 7) << 2;
#pragma unroll
      for (int i = 0; i < 4; ++i) {
        const int rr = r + 32 * i;
        const float4 v = *(const float4*)(Bm + (size_t)(col0 + rr) * ldb + kb + kq);
        *(float4*)(Bt + rr * LDSS + kq) = v;
      }
    } else {
      const int n  = t & 127;
      const int kh = (t >> 7) << 4;   // 0 or 16
#pragma unroll
      for (int kk = 0; kk < 16; ++kk) {
        const int k = kh + kk;
        Bt[n * LDSS + k] = Bm[(size_t)(kb + k) * ldb + col0 + n];
      }
    }
    __syncthreads();
    // --- 32 WMMAs over this K chunk (4 per k-step, 1:1 load:wmma) ---
#pragma unroll
    for (int k0 = 0; k0 < KT; k0 += 4) {
      const v2f a0 = *(const v2f*)(ap0 + k0);
      const v2f a1 = *(const v2f*)(ap1 + k0);
      const v2f b0 = *(const v2f*)(bp0 + k0);
      const v2f b1 = *(const v2f*)(bp1 + k0);
      acc[0][0] = wmma4(a0, b0, acc[0][0]);
      acc[0][1] = wmma4(a0, b1, acc[0][1]);
      acc[1][0] = wmma4(a1, b0, acc[1][0]);
      acc[1][1] = wmma4(a1, b1, acc[1][1]);
    }
  }

  // --- epilogue; C layout per ISA: lanes 0-15 -> M=v, lanes 16-31 -> M=v+8 ---
  const int mbase = (lane < 16) ? 0 : 8;
#pragma unroll
  for (int i = 0; i < 2; ++i) {
#pragma unroll
    for (int j = 0; j < 2; ++j) {
      const int col   = col0 + wn * 32 + j * 16 + m_l;
      const int rbase = row0 + wm * 32 + i * 16 + mbase;
#pragma unroll
      for (int v = 0; v < 8; ++v) {
        const int row = rbase + v;
        float x = acc[i][j][v];
        if (EPI == EPI_STORE) {
          C[(size_t)row * ldc + col] = x;
        } else if (EPI == EPI_RESID_BIAS) {
          x += resid[(size_t)row * ld_resid + col] + bias[col];
          C[(size_t)row * ldc + col] = x;
        } else {  // EPI_GATE
          x += bias[col];
          const float g  = 1.f / (1.f + __expf(-x));
          const float fg = catbuf[(size_t)row * ldcat + col];
          const float fs = catbuf[(size_t)row * ldcat + fs_off + col];
          C[(size_t)row * ldc + col]        = g * fg + (1.f - g) * fs;
          out_gate[(size_t)row * ldc + col] = g;
        }
      }
    }
  }
}

// ---------------------------------------------------------------------------
// bc[i] = bo[i] + sum_j Wo[i,j] * bv[j]   (D outputs; tiny)
// ---------------------------------------------------------------------------
__global__ __launch_bounds__(256)
void bias_combine(const float* __restrict__ Wo, const float* __restrict__ bv,
                  const float* __restrict__ bo, float* __restrict__ bc) {
  const int i = blockIdx.x * blockDim.x + threadIdx.x;
  if (i < DDIM) {
    float s = bo[i];
    const float* w = Wo + (size_t)i * DDIM;
    for (int j = 0; j < DDIM; ++j) s += w[j] * bv[j];
    bc[i] = s;
  }
}

// ---------------------------------------------------------------------------
// In-place LayerNorm over one row of length DDIM (256 thr x 4 elems).
// ---------------------------------------------------------------------------
__global__ __launch_bounds__(256)
void layernorm_rows(float* __restrict__ X, int ld,
                    const float* __restrict__ gamma,
                    const float* __restrict__ beta) {
  __shared__ float s_sum[8];
  __shared__ float s_sq[8];
  float* row = X + (size_t)blockIdx.x * ld;
  const int t = threadIdx.x;

  float4 x = *(const float4*)(row + t * 4);
  float s = x.x + x.y + x.z + x.w;
  float q = x.x * x.x + x.y * x.y + x.z * x.z + x.w * x.w;
#pragma unroll
  for (int off = 16; off > 0; off >>= 1) {
    s += __shfl_down(s, off);
    q += __shfl_down(q, off);
  }
  const int lane = t & 31, wv = t >> 5;
  if (lane == 0) { s_sum[wv] = s; s_sq[wv] = q; }
  __syncthreads();
  if (t == 0) {
    float ts = 0.f, tq = 0.f;
#pragma unroll
    for (int i = 0; i < 8; ++i) { ts += s_sum[i]; tq += s_sq[i]; }
    s_sum[0] = ts;
    s_sq[0]  = tq;
  }
  __syncthreads();
  const float mu  = s_sum[0] * (1.f / (float)DDIM);
  const float var = s_sq[0] * (1.f / (float)DDIM) - mu * mu;
  const float rs  = rsqrtf(var + LN_EPS);
  const float4 g  = *(const float4*)(gamma + t * 4);
  const float4 be = *(const float4*)(beta + t * 4);
  x.x = (x.x - mu) * rs * g.x + be.x;
  x.y = (x.y - mu) * rs * g.y + be.y;
  x.z = (x.z - mu) * rs * g.z + be.z;
  x.w = (x.w - mu) * rs * g.w + be.w;
  *(float4*)(row + t * 4) = x;
}

// ---------------------------------------------------------------------------
extern "C" void kernel_launch(void* const* d_in, const int* in_sizes, int n_in,
                              void* d_out, int out_size, void* d_ws, size_t ws_size,
                              hipStream_t stream) {
  (void)in_sizes; (void)n_in; (void)out_size; (void)ws_size;
  const int B = BDIM, D = DDIM;

  const float* g_emb  = (const float*)d_in[0];
  const float* s_emb  = (const float*)d_in[1];
  // d_in[2] = num_heads (unused: seq_len-1 attention is head-count invariant)
  const float* Wv_gs  = (const float*)d_in[5];
  const float* bv_gs  = (const float*)d_in[8];
  const float* Wo_gs  = (const float*)d_in[9];
  const float* bo_gs  = (const float*)d_in[10];
  const float* Wv_sg  = (const float*)d_in[13];
  const float* bv_sg  = (const float*)d_in[16];
  const float* Wo_sg  = (const float*)d_in[17];
  const float* bo_sg  = (const float*)d_in[18];
  const float* ln_g_g = (const float*)d_in[19];
  const float* ln_g_b = (const float*)d_in[20];
  const float* ln_s_g = (const float*)d_in[21];
  const float* ln_s_b = (const float*)d_in[22];
  const float* Wg     = (const float*)d_in[23];   // [D, 2D] row-major
  const float* bg     = (const float*)d_in[24];

  float* out_fused = (float*)d_out;                  // [B, D]
  float* out_gate  = out_fused + (size_t)B * D;      // [B, D]

  // workspace carve-out (~277 MB)
  float* cat   = (float*)d_ws;                       // [B, 2D] pre-LN then LN'd
  float* Wc_gs = cat + (size_t)B * 2 * D;            // [D, D] = Wo_gs @ Wv_gs
  float* Wc_sg = Wc_gs + (size_t)D * D;              // [D, D] = Wo_sg @ Wv_sg
  float* bc_gs = Wc_sg + (size_t)D * D;              // [D]
  float* bc_sg = bc_gs + D;                          // [D]

  const dim3 blk(256);

  // 1) combined biases
  bias_combine<<<dim3((D + 255) / 256), blk, 0, stream>>>(Wo_gs, bv_gs, bo_gs, bc_gs);
  bias_combine<<<dim3((D + 255) / 256), blk, 0, stream>>>(Wo_sg, bv_sg, bo_sg, bc_sg);

  // 2) combined weights: Wc = Wo @ Wv  (A=Wo, B'[k,n]=Wv[k,n] -> opB=N)
  {
    const dim3 grid(D / 128, D / 64);
    gemm_f32_wmma<EPI_STORE, false><<<grid, blk, 0, stream>>>(
        Wo_gs, D, Wv_gs, D, Wc_gs, D, D,
        nullptr, 0, nullptr, nullptr, 0, 0, nullptr);
    gemm_f32_wmma<EPI_STORE, false><<<grid, blk, 0, stream>>>(
        Wo_sg, D, Wv_sg, D, Wc_sg, D, D,
        nullptr, 0, nullptr, nullptr, 0, 0, nullptr);
  }

  // 3) attention GEMMs with residual+bias -> pre-LN halves of cat[B, 2D]
  {
    const dim3 grid(D / 128, B / 64);
    // graph half: g + s @ Wc_gs^T + bc_gs
    gemm_f32_wmma<EPI_RESID_BIAS, true><<<grid, blk, 0, stream>>>(
        s_emb, D, Wc_gs, D, cat, 2 * D, D,
        g_emb, D, bc_gs, nullptr, 0, 0, nullptr);
    // smiles half: s + g @ Wc_sg^T + bc_sg
    gemm_f32_wmma<EPI_RESID_BIAS, true><<<grid, blk, 0, stream>>>(
        g_emb, D, Wc_sg, D, cat + D, 2 * D, D,
        s_emb, D, bc_sg, nullptr, 0, 0, nullptr);
  }

  // 4) in-place LayerNorm on both halves
  layernorm_rows<<<dim3(B), blk, 0, stream>>>(cat,     2 * D, ln_g_g, ln_g_b);
  layernorm_rows<<<dim3(B), blk, 0, stream>>>(cat + D, 2 * D, ln_s_g, ln_s_b);

  // 5) gate GEMM (K = 2D) + sigmoid blend -> d_out
  {
    const dim3 grid(D / 128, B / 64);
    gemm_f32_wmma<EPI_GATE, true><<<grid, blk, 0, stream>>>(
        cat, 2 * D, Wg, 2 * D, out_fused, D, 2 * D,
        nullptr, 0, bg, cat, 2 * D, D, out_gate);
  }
}